// PermutoEncImpl_21242908246576
// MI455X (gfx1250) — compile-verified
//
#include <hip/hip_runtime.h>
#include <hip/hip_bf16.h>
#include <stdint.h>

#define N_LEVELS 16
#define HSIZE (1u << 18)
#define NPTS 262144
#define BLOCK 256
#define PTS_PER_BLOCK (BLOCK / N_LEVELS) /* 16 points, 48 floats staged per block */

typedef uint32_t u32;
typedef uint64_t u64;
typedef __attribute__((ext_vector_type(4))) u32 u32x4;
typedef __attribute__((ext_vector_type(8))) u32 u32x8;

__global__ __launch_bounds__(BLOCK) void permuto_enc_kernel(
    const float* __restrict__ positions,     // [NPTS,3]
    const float* __restrict__ lattice,       // [16, HSIZE, 2]
    const float* __restrict__ shifts,        // [16,3]
    const float* __restrict__ pos_scale,     // [3]
    float* __restrict__ out)                 // [NPTS, 32]
{
  __shared__ float s_pos[PTS_PER_BLOCK * 3];
  __shared__ float s_scale[N_LEVELS * 3];
  __shared__ float s_shift[N_LEVELS * 3];
  __shared__ float s_psc[3];

  const int t = threadIdx.x;

  // ---- stage small per-level constants (waves 1..) -------------------------
  if (t >= 32 && t < 32 + N_LEVELS * 3) s_shift[t - 32] = shifts[t - 32];
  if (t >= 96 && t < 99)                s_psc[t - 96]   = pos_scale[t - 96];
  if (t >= 128 && t < 128 + N_LEVELS) {
    // SCALE[l][i] = geomspace(16,512,16)[l] / sqrt((i+1)(i+2))
    //            = 2^(4+q) * base[rem][i]  with l = 3q+rem  (power-of-2 scale is EXACT)
    const int l   = t - 128;
    const int q   = l / 3;
    const int rem = l - 3 * q;
    const float m = (float)(1 << (4 + q));                 // 16..512, exact
    const float b0[3] = {0.70710678118654752f,             // 2^(-1/2)
                         0.89089871814033930f,             // 2^(-1/6)
                         1.12246204830937298f};            // 2^(1/6)
    const float b1[3] = {0.40824829046386302f,             // 1/sqrt6
                         0.51436061473901230f,             // 2^(1/3)/sqrt6
                         0.64805376576574020f};            // 2^(2/3)/sqrt6
    const float b2[3] = {0.28867513459481288f,             // 1/sqrt12
                         0.36370787865724040f,             // 2^(1/3)/sqrt12
                         0.45824321216203450f};            // 2^(2/3)/sqrt12
    s_scale[l * 3 + 0] = b0[rem] * m;
    s_scale[l * 3 + 1] = b1[rem] * m;
    s_scale[l * 3 + 2] = b2[rem] * m;
  }

  // ---- TDM: async tensor copy of this block's 16 positions into LDS -------
  // 1-D descriptor: 48 f32 elements, issued once by wave 0 (CDNA5 ISA ch.8).
  if (t < 32) {
    const u64 gaddr = (u64)(uintptr_t)positions +
                      (u64)blockIdx.x * (u64)(PTS_PER_BLOCK * 3 * sizeof(float));
    const u32 ldsa = (u32)(uintptr_t)(&s_pos[0]);
    u32x4 g0;
    g0[0] = 1u;                                    // count=1, user mode, no gather
    g0[1] = ldsa;                                  // lds_addr (bytes)
    g0[2] = (u32)gaddr;                            // global_addr[31:0]
    g0[3] = (u32)(gaddr >> 32) | (2u << 30);       // global_addr[56:32] | type=2
    u32x8 g1;
    g1[0] = (2u << 16);                            // mask=0, data_size=2 (4B), no flags
    g1[1] = (48u << 16);                           // tensor_dim0 = 48 (bits[79:48] lo)
    g1[2] = 0u;                                    // tensor_dim0 hi, tensor_dim1 lo
    g1[3] = (48u << 16);                           // tile_dim0 = 48 (bits[127:112])
    g1[4] = 0u;                                    // tile_dim1=0, tile_dim2=0
    g1[5] = 48u;                                   // tensor_dim0_stride = 48
    g1[6] = 0u;
    g1[7] = 0u;
    asm volatile("tensor_load_to_lds %0, %1" :: "s"(g0), "s"(g1) : "memory");
    __builtin_amdgcn_s_wait_tensorcnt(0);
  }
  __syncthreads();

  // ---- per-(point, level) lane ---------------------------------------------
  const int gtid  = blockIdx.x * BLOCK + t;
  const int level = gtid & (N_LEVELS - 1);
  const int n     = gtid >> 4;
  const int lp    = t >> 4;

  const float px = s_pos[lp * 3 + 0] * s_psc[0];
  const float py = s_pos[lp * 3 + 1] * s_psc[1];
  const float pz = s_pos[lp * 3 + 2] * s_psc[2];

  const float cf0 = (px + s_shift[level * 3 + 0]) * s_scale[level * 3 + 0];
  const float cf1 = (py + s_shift[level * 3 + 1]) * s_scale[level * 3 + 1];
  const float cf2 = (pz + s_shift[level * 3 + 2]) * s_scale[level * 3 + 2];

  // elevate to H_3 hyperplane in R^4 (same summation order as jnp.cumsum reversed)
  float ef[4];
  const float t12 = cf2 + cf1;
  ef[0] = t12 + cf0;
  ef[1] = t12 - cf0;
  ef[2] = cf2 - 2.0f * cf1;
  ef[3] = -(3.0f * cf2);

  // nearest remainder-0 lattice point (round to nearest multiple of 4, ties down)
  int r[4];
#pragma unroll
  for (int i = 0; i < 4; ++i) {
    const float v  = ef[i] * 0.25f;
    const float up = ceilf(v)  * 4.0f;
    const float dn = floorf(v) * 4.0f;
    r[i] = (int)(((up - ef[i]) < (ef[i] - dn)) ? up : dn);
  }
  const int ssum = (r[0] + r[1] + r[2] + r[3]) >> 2;   // exact: sum is multiple of 4

  // stable pairwise rank of residuals (uses pre-adjustment rem0)
  float diff[4];
#pragma unroll
  for (int i = 0; i < 4; ++i) diff[i] = ef[i] - (float)r[i];
  int rk[4];
#pragma unroll
  for (int a = 0; a < 4; ++a) {
    int c = 0;
#pragma unroll
    for (int b = 0; b < 4; ++b)
      c += (diff[b] > diff[a]) || ((diff[b] == diff[a]) && (b < a));
    rk[a] = c + ssum;
  }
  // single wrap correction
#pragma unroll
  for (int i = 0; i < 4; ++i) {
    if (rk[i] < 0)      { rk[i] += 4; r[i] += 4; }
    else if (rk[i] > 3) { rk[i] -= 4; r[i] -= 4; }
  }

  // barycentric weights (length-5 accumulator, static indexing via selects)
  float b5[5] = {0.f, 0.f, 0.f, 0.f, 0.f};
#pragma unroll
  for (int i = 0; i < 4; ++i) {
    const float del = (ef[i] - (float)r[i]) * 0.25f;
    const int ka = 3 - rk[i];
    const int kb = 4 - rk[i];
#pragma unroll
    for (int s = 0; s < 5; ++s)
      b5[s] += ((ka == s) ? del : 0.f) - ((kb == s) ? del : 0.f);
  }
  float w[4];
  w[0] = (b5[0] + 1.0f) + b5[4];
  w[1] = b5[1];
  w[2] = b5[2];
  w[3] = b5[3];

  // 4 simplex vertices: hash, gather (L2-resident table), weighted sum
  const float* __restrict__ tbl = lattice + (size_t)level * (size_t)(HSIZE * 2);
  float ax = 0.f, ay = 0.f;
#pragma unroll
  for (int kv = 0; kv < 4; ++kv) {
    const int thr = 3 - kv;
    const u32 c0 = (u32)(r[0] + kv - ((rk[0] > thr) ? 4 : 0));
    const u32 c1 = (u32)(r[1] + kv - ((rk[1] > thr) ? 4 : 0));
    const u32 c2 = (u32)(r[2] + kv - ((rk[2] > thr) ? 4 : 0));
    const u32 h  = (c0 * 1u) ^ (c1 * 2654435761u) ^ (c2 * 805459861u);
    const u32 idx = h & (HSIZE - 1u);
    const float2 f = *(const float2*)(tbl + (size_t)idx * 2u);
    ax += w[kv] * f.x;
    ay += w[kv] * f.y;
  }

  // coalesced non-temporal store: wave writes 256 contiguous bytes; NT keeps
  // the streamed 32MB output from evicting the hot 32MB table in L2
  union { float2 f2; unsigned long long u; } cv;
  cv.f2 = make_float2(ax, ay);
  unsigned long long* op = (unsigned long long*)out + ((size_t)n * N_LEVELS + (size_t)level);
  __builtin_nontemporal_store(cv.u, op);
}

extern "C" void kernel_launch(void* const* d_in, const int* in_sizes, int n_in,
                              void* d_out, int out_size, void* d_ws, size_t ws_size,
                              hipStream_t stream) {
  (void)in_sizes; (void)n_in; (void)out_size; (void)d_ws; (void)ws_size;
  const float* positions = (const float*)d_in[0];
  const float* lattice   = (const float*)d_in[1];
  const float* shifts    = (const float*)d_in[2];
  const float* pscale    = (const float*)d_in[3];
  float* out = (float*)d_out;

  const int total  = NPTS * N_LEVELS;          // one lane per (point, level)
  const int blocks = total / BLOCK;            // exact: 16384 blocks
  permuto_enc_kernel<<<dim3(blocks), dim3(BLOCK), 0, stream>>>(
      positions, lattice, shifts, pscale, out);
}